// CFConv_55121610276868
// MI455X (gfx1250) — compile-verified
//
#include <hip/hip_runtime.h>

typedef __attribute__((ext_vector_type(2))) float v2f;
typedef __attribute__((ext_vector_type(8))) float v8f;

#define NUM_ATOMS 40000
#define NUM_EDGES 640000
#define NFM 128

// ---------------------------------------------------------------------------
// WMMA fp32 GEMM:  C[M x 128] = A[M x 128] @ W[128 x 128]  (+ optional bias)
// One wave per 16x16 output tile; 8 waves/block cover 16 rows x 128 cols.
// V_WMMA_F32_16X16X4_F32, K=128 in steps of 4 (32 WMMAs per tile).
//
// VGPR layouts (ISA 7.12.2):
//   A (16x4 f32): lane L, M = L&15; lanes 0-15 hold K=k+0/k+1 (v0/v1);
//                 lanes 16-31 hold K=k+2/k+3.
//   B (4x16 f32): mirrored: N = L&15; same K split per half-wave.
//   C/D (16x16 f32, 8 VGPRs): vgpr r: lanes 0-15 -> M=r, lanes 16-31 -> M=r+8;
//                 N = L&15.
// ---------------------------------------------------------------------------
__global__ __launch_bounds__(256)
void cfconv_gemm_wmma_f32(const float* __restrict__ A,
                          const float* __restrict__ W,
                          const float* __restrict__ bias,
                          float* __restrict__ C)
{
    const int wave  = threadIdx.x >> 5;     // 0..7 -> N tile
    const int lane  = threadIdx.x & 31;
    const int l15   = lane & 15;
    const int hi    = lane >> 4;            // 0 or 1
    const int tileM = blockIdx.x;           // 16-row stripe

    const int row   = tileM * 16 + l15;     // A row this lane reads
    const int col   = wave * 16 + l15;      // B/C column this lane owns
    const int khalf = hi * 2;               // K sub-offset for this half-wave

    const float* Arow = A + (size_t)row * NFM;

    v8f acc = {};
#pragma unroll 4
    for (int k = 0; k < NFM; k += 4) {
        const int kb = k + khalf;
        v2f a, b;
        a.x = Arow[kb + 0];
        a.y = Arow[kb + 1];
        b.x = W[(size_t)(kb + 0) * NFM + col];
        b.y = W[(size_t)(kb + 1) * NFM + col];
        acc = __builtin_amdgcn_wmma_f32_16x16x4_f32(
            /*neg_a=*/false, a, /*neg_b=*/false, b,
            /*c_mod=*/(short)0, acc, /*reuse_a=*/false, /*reuse_b=*/false);
    }

    const float bv = bias ? bias[col] : 0.0f;
    const int rbase = tileM * 16 + hi * 8;
#pragma unroll
    for (int r = 0; r < 8; ++r) {
        C[(size_t)(rbase + r) * NFM + col] = acc[r] + bv;
    }
}

// ---------------------------------------------------------------------------
// Edge phase: conv[i, :] = sum_{e : seg_i[e]==i} w_ij[e, :] * f[idx_j[e], :]
//
// seg_i is sorted -> each atom's edges are one contiguous range (binary search
// once per block; no atomics). Bandwidth shape: one block per atom, 4 waves;
// each lane owns 4 contiguous channels (float4), so one edge row (512 B) is
// exactly one global_load_b128 per lane across a wave. Wave w walks edges
// start+w, start+w+4, ... (4 edge rows in flight per loop trip); the 4 wave
// partials are combined through LDS and stored with a single b128 per lane.
// w_ij streams once from HBM (327 MB, the roofline term); f (20.5 MB) gathers
// out of the 192 MB L2.
// ---------------------------------------------------------------------------
__global__ __launch_bounds__(128)
void cfconv_edge_segsum(const float* __restrict__ f,
                        const float* __restrict__ w_ij,
                        const int*   __restrict__ seg_i,
                        const int*   __restrict__ idx_j,
                        float* __restrict__ conv)
{
    const int atom = blockIdx.x;
    const int lane = threadIdx.x & 31;   // channel group (4 channels each)
    const int wave = threadIdx.x >> 5;   // 0..3 -> edge interleave
    const int c4   = lane * 4;

    __shared__ int   s_bounds[2];
    __shared__ float4 s_part[4][32];     // 2 KB

    if (threadIdx.x == 0) {
        // lower_bound(seg_i, atom)
        int lo = 0, hi = NUM_EDGES;
        while (lo < hi) {
            int mid = (lo + hi) >> 1;
            if (seg_i[mid] < atom) lo = mid + 1; else hi = mid;
        }
        s_bounds[0] = lo;
        // upper_bound(seg_i, atom)
        hi = NUM_EDGES;
        while (lo < hi) {
            int mid = (lo + hi) >> 1;
            if (seg_i[mid] <= atom) lo = mid + 1; else hi = mid;
        }
        s_bounds[1] = lo;
    }
    __syncthreads();

    const int start = s_bounds[0];
    const int end   = s_bounds[1];

    float4 acc = make_float4(0.f, 0.f, 0.f, 0.f);
    for (int e = start + wave; e < end; e += 4) {
        // stream w_ij ahead of use (global_prefetch_b8 on gfx1250)
        if (e + 8 < end) {
            __builtin_prefetch(&w_ij[(size_t)(e + 8) * NFM + c4], 0, 0);
        }
        const int j = idx_j[e];
        const float4 w  = *(const float4*)&w_ij[(size_t)e * NFM + c4];
        const float4 fv = *(const float4*)&f[(size_t)j * NFM + c4];
        acc.x = fmaf(w.x, fv.x, acc.x);
        acc.y = fmaf(w.y, fv.y, acc.y);
        acc.z = fmaf(w.z, fv.z, acc.z);
        acc.w = fmaf(w.w, fv.w, acc.w);
    }

    s_part[wave][lane] = acc;
    __syncthreads();

    if (wave == 0) {
        const float4 p0 = s_part[0][lane];
        const float4 p1 = s_part[1][lane];
        const float4 p2 = s_part[2][lane];
        const float4 p3 = s_part[3][lane];
        float4 r;
        r.x = (p0.x + p1.x) + (p2.x + p3.x);
        r.y = (p0.y + p1.y) + (p2.y + p3.y);
        r.z = (p0.z + p1.z) + (p2.z + p3.z);
        r.w = (p0.w + p1.w) + (p2.w + p3.w);
        *(float4*)&conv[(size_t)atom * NFM + c4] = r;
    }
}

// ---------------------------------------------------------------------------
// Launch: f = x @ W_in ; conv = edge-segsum ; out = conv @ W_out + b_out
// Workspace: f (20.48 MB) then conv (20.48 MB).
// ---------------------------------------------------------------------------
extern "C" void kernel_launch(void* const* d_in, const int* in_sizes, int n_in,
                              void* d_out, int out_size, void* d_ws, size_t ws_size,
                              hipStream_t stream)
{
    const float* x     = (const float*)d_in[0];   // (40000, 128)
    const float* w_ij  = (const float*)d_in[1];   // (640000, 128)
    const int*   seg_i = (const int*)  d_in[2];   // (640000,)
    const int*   idx_j = (const int*)  d_in[3];   // (640000,)
    // d_in[4] = seg_i_sum scalar (== NUM_ATOMS), compile-time constant here
    const float* W_in  = (const float*)d_in[5];   // (128, 128)
    const float* W_out = (const float*)d_in[6];   // (128, 128)
    const float* b_out = (const float*)d_in[7];   // (128,)

    float* out  = (float*)d_out;                  // (40000, 128)
    float* f    = (float*)d_ws;                                   // 40000*128 f32
    float* conv = (float*)((char*)d_ws + (size_t)NUM_ATOMS * NFM * sizeof(float));

    const int mTiles = NUM_ATOMS / 16;            // 2500, exact

    // f = x @ W_in (no bias)
    cfconv_gemm_wmma_f32<<<mTiles, 256, 0, stream>>>(x, W_in, nullptr, f);

    // conv[i] = sum_e w_ij[e] * f[idx_j[e]]  (seg_i sorted, no atomics)
    cfconv_edge_segsum<<<NUM_ATOMS, 128, 0, stream>>>(f, w_ij, seg_i, idx_j, conv);

    // out = conv @ W_out + b_out
    cfconv_gemm_wmma_f32<<<mTiles, 256, 0, stream>>>(conv, W_out, b_out, out);
}